// GCN_16724602650711
// MI455X (gfx1250) — compile-verified
//
#include <hip/hip_runtime.h>
#include <hip/hip_bf16.h>
#include <math.h>

// -------- types --------
typedef float v2f __attribute__((ext_vector_type(2)));
typedef float v8f __attribute__((ext_vector_type(8)));

#define BN_EPS 1e-5f

// ============================================================
// Graph normalization precompute
// ============================================================
__global__ void deg_init_kernel(float* deg, int n) {
  int i = blockIdx.x * blockDim.x + threadIdx.x;
  if (i < n) deg[i] = 1.0f;               // self-loop contributes 1
}

__global__ void deg_acc_kernel(const int* __restrict__ dst, float* deg, int E) {
  int e = blockIdx.x * blockDim.x + threadIdx.x;
  if (e < E) atomicAdd(&deg[dst[e]], 1.0f);
}

__global__ void dinv_kernel(const float* __restrict__ deg, float* dinv, int n) {
  int i = blockIdx.x * blockDim.x + threadIdx.x;
  if (i < n) dinv[i] = rsqrtf(fmaxf(deg[i], 1.0f));
}

__global__ void norm_kernel(const int* __restrict__ src, const int* __restrict__ dst,
                            const float* __restrict__ dinv, float* nrm, int E) {
  int e = blockIdx.x * blockDim.x + threadIdx.x;
  if (e < E) nrm[e] = dinv[src[e]] * dinv[dst[e]];
}

// ============================================================
// WMMA f32 GEMM, register-blocked on N:
//   one wave computes a 16 x (16*NT) strip of C = A[M x K] * B[K x Nd]
//   A fragment loaded once per k-step, reused across NT B fragments.
// M % 16 == 0, Nd % (16*NT) == 0, K % 4 == 0
// ============================================================
template <int NT>
__global__ void gemm_wmma_f32_nt(const float* __restrict__ A,
                                 const float* __restrict__ B,
                                 float* __restrict__ C,
                                 int M, int K, int Nd) {
  int wavesPerBlock = blockDim.x >> 5;
  int wave = blockIdx.x * wavesPerBlock + ((int)threadIdx.x >> 5);
  int nblk = (Nd >> 4) / NT;
  int mtiles = M >> 4;
  if (wave >= mtiles * nblk) return;   // wave-uniform: EXEC stays all-ones
  int tm = wave / nblk;
  int tb = wave - tm * nblk;
  int lane = (int)threadIdx.x & 31;
  int half = lane >> 4;                // 0: K pair {0,1}, 1: K pair {2,3}
  int l = lane & 15;

  const float* arow  = A + (size_t)(tm * 16 + l) * K;   // A: M on lanes
  const float* bbase = B + (tb * (16 * NT) + l);        // B: N on lanes

  const v8f vzero = {0.f, 0.f, 0.f, 0.f, 0.f, 0.f, 0.f, 0.f};
  v8f c[NT];
#pragma unroll
  for (int j = 0; j < NT; ++j) c[j] = vzero;

  for (int kb = 0; kb < K; kb += 4) {
    int ka = kb + 2 * half;
    v2f a;
    a.x = arow[ka];
    a.y = arow[ka + 1];
#pragma unroll
    for (int j = 0; j < NT; ++j) {
      v2f b;
      b.x = bbase[(size_t)ka * Nd + j * 16];
      b.y = bbase[(size_t)(ka + 1) * Nd + j * 16];
      c[j] = __builtin_amdgcn_wmma_f32_16x16x4_f32(
          /*neg_a=*/false, a, /*neg_b=*/false, b,
          /*c_mod=*/(short)0, c[j], /*reuse_a=*/false, /*reuse_b=*/false);
    }
  }

  int rbase = tm * 16 + half * 8;      // lanes 16-31 hold rows M+8..M+15
#pragma unroll
  for (int j = 0; j < NT; ++j) {
    int cbase = tb * (16 * NT) + j * 16 + l;
#pragma unroll
    for (int v = 0; v < 8; ++v)
      C[(size_t)(rbase + v) * Nd + cbase] = c[j][v];
  }
}

// ============================================================
// GCN scatter: out = bias + selfloop + sum_edges
// hlin has leading dim ldh (>= Nd, for the padded last layer)
// ============================================================
__global__ void scatter_init_kernel(const float* __restrict__ hlin,
                                    const float* __restrict__ bias,
                                    const float* __restrict__ dinv,
                                    float* __restrict__ out,
                                    int tot, int Nd, int ldh) {
  int idx = blockIdx.x * blockDim.x + threadIdx.x;
  if (idx >= tot) return;
  int n = idx / Nd;
  int f = idx - n * Nd;
  float dv = dinv[n];
  out[idx] = bias[f] + hlin[(size_t)n * ldh + f] * dv * dv;  // self-loop norm = dinv^2
}

// one wave per edge: per-edge scalars loaded once, lanes stride the features
__global__ void scatter_edges_kernel(const float* __restrict__ hlin,
                                     const int* __restrict__ src,
                                     const int* __restrict__ dst,
                                     const float* __restrict__ nrm,
                                     float* __restrict__ out,
                                     int E, int Nd, int ldh) {
  int wave = blockIdx.x * ((int)blockDim.x >> 5) + ((int)threadIdx.x >> 5);
  if (wave >= E) return;               // wave-uniform
  int lane = (int)threadIdx.x & 31;
  int s = src[wave];
  int d = dst[wave];
  float w = nrm[wave];
  const float* hs = hlin + (size_t)s * ldh;
  float* od = out + (size_t)d * Nd;
  for (int f = lane; f < Nd; f += 32)
    atomicAdd(&od[f], hs[f] * w);
}

// ============================================================
// BatchNorm stats: one block per feature column
// ============================================================
__global__ void bn_stats_kernel(const float* __restrict__ h,
                                float* __restrict__ mu, float* __restrict__ rvar,
                                int Nrows, int Nd) {
  int j = blockIdx.x;
  __shared__ float ssum[256];
  __shared__ float ssq[256];
  float s = 0.f, q = 0.f;
  for (int n = threadIdx.x; n < Nrows; n += blockDim.x) {
    float v = h[(size_t)n * Nd + j];
    s += v;
    q += v * v;
  }
  ssum[threadIdx.x] = s;
  ssq[threadIdx.x] = q;
  __syncthreads();
  for (int st = (int)blockDim.x >> 1; st > 0; st >>= 1) {
    if ((int)threadIdx.x < st) {
      ssum[threadIdx.x] += ssum[threadIdx.x + st];
      ssq[threadIdx.x] += ssq[threadIdx.x + st];
    }
    __syncthreads();
  }
  if (threadIdx.x == 0) {
    float m = ssum[0] / (float)Nrows;
    float var = ssq[0] / (float)Nrows - m * m;   // biased var
    mu[j] = m;
    rvar[j] = rsqrtf(var + BN_EPS);
  }
}

// fused BN apply + exact-erf GELU, in place
__global__ void bn_gelu_kernel(float* __restrict__ h,
                               const float* __restrict__ mu, const float* __restrict__ rvar,
                               const float* __restrict__ g, const float* __restrict__ be,
                               int tot, int Nd) {
  int idx = blockIdx.x * blockDim.x + threadIdx.x;
  if (idx >= tot) return;
  int j = idx % Nd;
  float y = g[j] * (h[idx] - mu[j]) * rvar[j] + be[j];
  h[idx] = 0.5f * y * (1.0f + erff(y * 0.70710678118654752f));
}

// zero-pad W[K x Nd] -> Wp[K x Ndp]
__global__ void pad_w_kernel(const float* __restrict__ W, float* __restrict__ Wp,
                             int K, int Nd, int Ndp) {
  int idx = blockIdx.x * blockDim.x + threadIdx.x;
  int tot = K * Ndp;
  if (idx >= tot) return;
  int k = idx / Ndp;
  int j = idx - k * Ndp;
  Wp[idx] = (j < Nd) ? W[k * Nd + j] : 0.0f;
}

// per-row log_softmax over 8 classes
__global__ void logsoftmax8_kernel(const float* __restrict__ h, float* __restrict__ out, int Nrows) {
  int n = blockIdx.x * blockDim.x + threadIdx.x;
  if (n >= Nrows) return;
  float v[8];
  float mx = -1e30f;
#pragma unroll
  for (int c = 0; c < 8; ++c) { v[c] = h[n * 8 + c]; mx = fmaxf(mx, v[c]); }
  float s = 0.f;
#pragma unroll
  for (int c = 0; c < 8; ++c) s += expf(v[c] - mx);
  float ls = logf(s);
#pragma unroll
  for (int c = 0; c < 8; ++c) out[n * 8 + c] = v[c] - mx - ls;
}

// ============================================================
static inline int cdiv(int a, int b) { return (a + b - 1) / b; }

extern "C" void kernel_launch(void* const* d_in, const int* in_sizes, int n_in,
                              void* d_out, int out_size, void* d_ws, size_t ws_size,
                              hipStream_t stream) {
  (void)n_in; (void)out_size; (void)ws_size;
  const int dims[8] = {128, 16, 32, 64, 128, 256, 512, 8};

  const float* x = (const float*)d_in[0];
  const int* eidx = (const int*)d_in[1];
  const int Nn = in_sizes[0] / dims[0];     // 20000
  const int Ee = in_sizes[1] / 2;           // 320000
  const int* src = eidx;
  const int* dst = eidx + Ee;

  const float* Ws[7]; const float* bs[7];
  const float* gs[6]; const float* bes[6];
  for (int i = 0; i < 7; ++i) { Ws[i] = (const float*)d_in[2 + i]; bs[i] = (const float*)d_in[9 + i]; }
  for (int i = 0; i < 6; ++i) { gs[i] = (const float*)d_in[16 + i]; bes[i] = (const float*)d_in[22 + i]; }

  // workspace carve-up
  char* wsb = (char*)d_ws;
  size_t off = 0;
  auto carve = [&](size_t bytes) -> void* {
    void* p = wsb + off;
    off += (bytes + 255) & ~(size_t)255;
    return p;
  };
  float* bufA = (float*)carve((size_t)Nn * 512 * sizeof(float));  // hlin
  float* bufB = (float*)carve((size_t)Nn * 512 * sizeof(float));  // layer output / h
  float* dinv = (float*)carve((size_t)Nn * sizeof(float));
  float* deg  = (float*)carve((size_t)Nn * sizeof(float));
  float* nrm  = (float*)carve((size_t)Ee * sizeof(float));
  float* mu   = (float*)carve(512 * sizeof(float));
  float* rvar = (float*)carve(512 * sizeof(float));
  float* wpad = (float*)carve(512 * 16 * sizeof(float));

  const int T = 256;                       // 8 waves/block
  const int WPB = T >> 5;

  // graph normalization
  deg_init_kernel<<<cdiv(Nn, T), T, 0, stream>>>(deg, Nn);
  deg_acc_kernel<<<cdiv(Ee, T), T, 0, stream>>>(dst, deg, Ee);
  dinv_kernel<<<cdiv(Nn, T), T, 0, stream>>>(deg, dinv, Nn);
  norm_kernel<<<cdiv(Ee, T), T, 0, stream>>>(src, dst, dinv, nrm, Ee);

  const float* h = x;
  for (int i = 0; i < 7; ++i) {
    const int K = dims[i];
    const int Nd = dims[i + 1];
    const int Ndp = (Nd < 16) ? 16 : Nd;   // pad last layer (dout=8) to 16

    const float* Wuse = Ws[i];
    if (Ndp != Nd) {
      pad_w_kernel<<<cdiv(K * Ndp, T), T, 0, stream>>>(Ws[i], wpad, K, Nd, Ndp);
      Wuse = wpad;
    }

    // hlin = h @ W  (fp32 WMMA, N-register-blocked)
    const int nt = (Ndp >= 64) ? 4 : ((Ndp >= 32) ? 2 : 1);
    const int waves = (Nn >> 4) * ((Ndp >> 4) / nt);
    if (nt == 4)
      gemm_wmma_f32_nt<4><<<cdiv(waves, WPB), T, 0, stream>>>(h, Wuse, bufA, Nn, K, Ndp);
    else if (nt == 2)
      gemm_wmma_f32_nt<2><<<cdiv(waves, WPB), T, 0, stream>>>(h, Wuse, bufA, Nn, K, Ndp);
    else
      gemm_wmma_f32_nt<1><<<cdiv(waves, WPB), T, 0, stream>>>(h, Wuse, bufA, Nn, K, Ndp);

    // out = bias + selfloop + edge scatter (one wave per edge)
    scatter_init_kernel<<<cdiv(Nn * Nd, T), T, 0, stream>>>(bufA, bs[i], dinv, bufB, Nn * Nd, Nd, Ndp);
    scatter_edges_kernel<<<cdiv(Ee, WPB), T, 0, stream>>>(bufA, src, dst, nrm, bufB, Ee, Nd, Ndp);

    if (i < 6) {
      bn_stats_kernel<<<Nd, T, 0, stream>>>(bufB, mu, rvar, Nn, Nd);
      bn_gelu_kernel<<<cdiv(Nn * Nd, T), T, 0, stream>>>(bufB, mu, rvar, gs[i], bes[i], Nn * Nd, Nd);
    }
    h = bufB;
  }

  logsoftmax8_kernel<<<cdiv(Nn, T), T, 0, stream>>>(bufB, (float*)d_out, Nn);
}